// KAN1D_62328565399938
// MI455X (gfx1250) — compile-verified
//
#include <hip/hip_runtime.h>
#include <math.h>

typedef float v2f __attribute__((ext_vector_type(2)));
typedef float v8f __attribute__((ext_vector_type(8)));

#define KNOTS 512
#define LN_EPS 1e-5f

#if defined(__HIP_DEVICE_COMPILE__)
  #if __has_builtin(__builtin_amdgcn_wmma_f32_16x16x4_f32)
    #define HAVE_WMMA_F32X4 1
  #endif
  #if __has_builtin(__builtin_amdgcn_global_load_async_to_lds_b32) && \
      __has_builtin(__builtin_amdgcn_s_wait_asynccnt)
    #define HAVE_ASYNC_LDS 1
  #endif
#endif

// ---------------------------------------------------------------------------
// Kernel 1: S_gw = sum_k gamma[k]*w[k], S_bw = sum_k beta[k]*w[k]  (K = 512).
// One wave32. Reduction done with V_WMMA_F32_16X16X4_F32: A holds 64 products
// per step, B = all-ones => D[m][n] = row-sum of A; accumulate C over 8 steps.
// Total = (sum of 8 D VGPRs on lane 0) + (same on lane 16)  [16x16 f32 D layout].
// Exact f32 arithmetic (WMMA f32 path), no precision loss vs the reference.
// ---------------------------------------------------------------------------
__global__ __launch_bounds__(32) void kan_reduce(const float* __restrict__ gamma,
                                                 const float* __restrict__ beta,
                                                 const float* __restrict__ w,
                                                 float* __restrict__ ws) {
  const int lane = threadIdx.x;
#ifdef HAVE_WMMA_F32X4
  const int m  = lane & 15;
  const int kk = (lane < 16) ? 0 : 2;   // A 16x4 layout: lanes 16-31 hold K=2,3
  v8f accg = {0.f,0.f,0.f,0.f,0.f,0.f,0.f,0.f};
  v8f accb = {0.f,0.f,0.f,0.f,0.f,0.f,0.f,0.f};
  v2f ones; ones.x = 1.0f; ones.y = 1.0f;
  for (int c = 0; c < 8; ++c) {
    const int base = c * 64 + m * 4 + kk;      // bijection onto [c*64, c*64+64)
    const float w0 = w[base], w1 = w[base + 1];
    v2f ag, ab;
    ag.x = gamma[base] * w0;  ag.y = gamma[base + 1] * w1;
    ab.x = beta[base]  * w0;  ab.y = beta[base + 1]  * w1;
    // 8 args: (neg_a, A, neg_b, B, c_mod, C, reuse_a, reuse_b)
    accg = __builtin_amdgcn_wmma_f32_16x16x4_f32(false, ag, false, ones,
                                                 (short)0, accg, false, false);
    accb = __builtin_amdgcn_wmma_f32_16x16x4_f32(false, ab, false, ones,
                                                 (short)0, accb, false, false);
  }
  float sg = 0.f, sb = 0.f;
  #pragma unroll
  for (int r = 0; r < 8; ++r) { sg += accg[r]; sb += accb[r]; }
  // D col N=0 lives on lane 0 (M=0..7) and lane 16 (M=8..15)
  float tg = __shfl(sg, 0, 32) + __shfl(sg, 16, 32);
  float tb = __shfl(sb, 0, 32) + __shfl(sb, 16, 32);
#else
  float sg = 0.f, sb = 0.f;
  for (int k = lane; k < KNOTS; k += 32) {
    const float wk = w[k];
    sg += gamma[k] * wk;
    sb += beta[k]  * wk;
  }
  for (int off = 16; off > 0; off >>= 1) {
    sg += __shfl_xor(sg, off, 32);
    sb += __shfl_xor(sb, off, 32);
  }
  float tg = sg, tb = sb;
#endif
  if (lane == 0) { ws[0] = tg; ws[1] = tb; }
}

// ---------------------------------------------------------------------------
// Kernel 2: closed-form spline + LayerNorm + head, elementwise over N.
// gw table staged into LDS via gfx1250 async global->LDS loads.
// ---------------------------------------------------------------------------
__global__ __launch_bounds__(256) void kan_main(const float* __restrict__ x,
                                                const float* __restrict__ gamma,
                                                const float* __restrict__ hw,
                                                const float* __restrict__ hb,
                                                const float* __restrict__ ws,
                                                float* __restrict__ out,
                                                int n) {
  __shared__ float s_g[KNOTS];
  __shared__ float s_w[KNOTS];
  __shared__ float s_gw[KNOTS];

#ifdef HAVE_ASYNC_LDS
  for (int k = threadIdx.x; k < KNOTS; k += 256) {
    __builtin_amdgcn_global_load_async_to_lds_b32(
        (__attribute__((address_space(1))) int*)(gamma + k),
        (__attribute__((address_space(3))) int*)(&s_g[k]), 0, 0);
    __builtin_amdgcn_global_load_async_to_lds_b32(
        (__attribute__((address_space(1))) int*)(hw + k),
        (__attribute__((address_space(3))) int*)(&s_w[k]), 0, 0);
  }
  __builtin_amdgcn_s_wait_asynccnt(0);
#else
  for (int k = threadIdx.x; k < KNOTS; k += 256) {
    s_g[k] = gamma[k];
    s_w[k] = hw[k];
  }
#endif
  __syncthreads();
  for (int k = threadIdx.x; k < KNOTS; k += 256) s_gw[k] = s_g[k] * s_w[k];
  __syncthreads();

  const float S_gw = ws[0];
  const float S_bw = ws[1];
  const float bias = hb[0];
  const float cK   = (float)KNOTS;
  const float invK = 1.0f / cK;
  const float denom = (3.0f - (-3.0f)) + 1e-8f;   // == 6.0f in f32, as in reference
  const int stride = gridDim.x * blockDim.x;

  for (int i = blockIdx.x * blockDim.x + threadIdx.x; i < n; i += stride) {
    const float xv = x[i];
    const float x0 = (xv + 3.0f) / denom;
    float u = fmodf(x0 * cK, cK);                 // periodic wrap, python-mod sign
    u = (u < 0.f) ? (u + cK) : u;
    const float fi = floorf(u);
    const float t  = u - fi;
    const int base = (int)fi;

    const float t2 = t * t, t3 = t2 * t;
    const float omt = 1.0f - t;
    const float b0 = omt * omt * omt * (1.0f / 6.0f);
    const float b1 = (3.0f * t3 - 6.0f * t2 + 4.0f) * (1.0f / 6.0f);
    const float b2 = (-3.0f * t3 + 3.0f * t2 + 3.0f * t + 1.0f) * (1.0f / 6.0f);
    const float b3 = t3 * (1.0f / 6.0f);

    const float mu = (b0 + b1 + b2 + b3) * invK;
    const float d0 = b0 - mu, d1 = b1 - mu, d2 = b2 - mu, d3 = b3 - mu;
    const float var = (d0 * d0 + d1 * d1 + d2 * d2 + d3 * d3 +
                       (cK - 4.0f) * (mu * mu)) * invK;
    const float rstd = rsqrtf(var + LN_EPS);

    const int i0 = base & (KNOTS - 1);
    const int i1 = (base + 1) & (KNOTS - 1);
    const int i2 = (base + 2) & (KNOTS - 1);
    const int i3 = (base + 3) & (KNOTS - 1);
    const float dot = b0 * s_gw[i0] + b1 * s_gw[i1] +
                      b2 * s_gw[i2] + b3 * s_gw[i3];

    out[i] = rstd * (dot - mu * S_gw) + S_bw + bias;
  }
}

extern "C" void kernel_launch(void* const* d_in, const int* in_sizes, int n_in,
                              void* d_out, int out_size, void* d_ws, size_t ws_size,
                              hipStream_t stream) {
  const float* x     = (const float*)d_in[0];   // (N,1) f32
  const float* gamma = (const float*)d_in[1];   // (K,)  f32
  const float* beta  = (const float*)d_in[2];   // (K,)  f32
  const float* hw    = (const float*)d_in[3];   // (1,K) f32
  const float* hb    = (const float*)d_in[4];   // (1,)  f32
  float* out = (float*)d_out;
  float* ws  = (float*)d_ws;
  const int n = in_sizes[0];

  kan_reduce<<<1, 32, 0, stream>>>(gamma, beta, hw, ws);
  kan_main<<<256, 256, 0, stream>>>(x, gamma, hw, hb, ws, out, n);
}